// DestSelectionPolicy_61512521613575
// MI455X (gfx1250) — compile-verified
//
#include <hip/hip_runtime.h>
#include <hip/hip_bf16.h>
#include <stdint.h>

#define NODE_DIM 64
#define EPS 1e-16f

typedef __attribute__((ext_vector_type(2))) float v2f;
typedef __attribute__((ext_vector_type(8))) float v8f;

// ---------------------------------------------------------------------------
// Kernel 1: zero-init seg_max / seg_sum (n*2 floats each, contiguous).
// ---------------------------------------------------------------------------
__global__ void init_seg_kernel(float* __restrict__ seg, int count) {
    int i = blockIdx.x * blockDim.x + threadIdx.x;
    if (i < count) seg[i] = 0.0f;
}

// ---------------------------------------------------------------------------
// Kernel 2: per-node projection P = X(16x64 tile) @ B(64x16) via
// V_WMMA_F32_16X16X4_F32 (fp32 WMMA, K-loop of 16 steps).
// B columns: 0 -> W[0,0:64], 1 -> W[1,0:64], 2 -> W[0,64:128], 3 -> W[1,64:128],
// columns 4..15 are zero.  One wave handles one 16-node tile.
//
// fp32 WMMA VGPR layouts (ISA 7.12.2):
//   A 16x4 : lane l holds row M=l&15; VGPR0/1 = K{0,1} (lanes 0-15) or K{2,3}
//   B 4x16 : lane l holds col N=l&15; VGPR0/1 = K{0,1} (lanes 0-15) or K{2,3}
//   C 16x16: lane l holds col N=l&15; VGPR v = row M = v + (l>=16 ? 8 : 0)
//
// The B fragment is loop-invariant: preload all 16 steps (32 VGPRs) before the
// K-loop with UNCONDITIONAL loads (safe: wbase==0 for invalid lanes keeps the
// index within W's 256 elements) + v_cndmask zeroing — no exec-mask branches.
// ---------------------------------------------------------------------------
__global__ void node_proj_wmma_kernel(const float* __restrict__ x,
                                      const float* __restrict__ W,
                                      float* __restrict__ p_row,
                                      float* __restrict__ p_col,
                                      int n_nodes) {
    const int lane = threadIdx.x & 31;
    const int wave = threadIdx.x >> 5;
    const int tile = blockIdx.x * (blockDim.x >> 5) + wave;
    const int row_base = tile * 16;
    if (row_base >= n_nodes) return;   // wave-uniform: EXEC stays all-1s for WMMA

    const int M    = lane & 15;        // A-row for this lane (also B/C column id)
    const int half = lane >> 4;        // 0: K{0,1}/rows 0-7, 1: K{2,3}/rows 8-15
    const int koff = half * 2;

    // B-matrix source: Bmat[k][N] = W[(N&1)*128 + (N>>1)*64 + k] for N<4, else 0
    const bool bvalid = (M < 4);
    const int  wbase  = bvalid ? ((M & 1) * 128 + (M >> 1) * 64) : 0;

    // Preload entire B fragment (loop-invariant), unconditional loads + select.
    v2f bfrag[16];
#pragma unroll
    for (int s = 0; s < 16; ++s) {
        const int kk = 4 * s + koff;
        const float w0 = W[wbase + kk];
        const float w1 = W[wbase + kk + 1];
        bfrag[s].x = bvalid ? w0 : 0.0f;
        bfrag[s].y = bvalid ? w1 : 0.0f;
    }

    const float* __restrict__ xrow = x + (size_t)(row_base + M) * NODE_DIM;

    v8f acc = {};
#pragma unroll
    for (int s = 0; s < 16; ++s) {
        const int kk = 4 * s + koff;
        // 8-byte aligned (kk is even, xrow is 256B aligned): one global_load_b64
        const v2f a = *(const v2f*)(xrow + kk);
        acc = __builtin_amdgcn_wmma_f32_16x16x4_f32(
            /*neg_a=*/false, a, /*neg_b=*/false, bfrag[s],
            /*c_mod=*/(short)0, acc, /*reuse_a=*/false, /*reuse_b=*/false);
    }

    // Lanes whose column N = M < 4 own the useful outputs.
    if (bvalid) {
        const int c = M;                     // output column 0..3
#pragma unroll
        for (int m = 0; m < 8; ++m) {
            const int node = row_base + half * 8 + m;
            const float v = acc[m];
            if (c < 2) p_row[node * 2 + c]       = v;
            else       p_col[node * 2 + (c - 2)] = v;
        }
    }
}

// ---------------------------------------------------------------------------
// Per-edge score helper: s[c] = relu(p_row[r][c] + p_col[c_node][c] + b[c])
// ---------------------------------------------------------------------------
__device__ __forceinline__ void edge_scores(const long long* __restrict__ ei,
                                            const float* __restrict__ p_row,
                                            const float* __restrict__ p_col,
                                            const float* __restrict__ b,
                                            int e, int num_edges,
                                            int& r, float& s0, float& s1) {
    r = (int)ei[e];
    const int c = (int)ei[num_edges + e];
    s0 = fmaxf(p_row[r * 2 + 0] + p_col[c * 2 + 0] + b[0], 0.0f);
    s1 = fmaxf(p_row[r * 2 + 1] + p_col[c * 2 + 1] + b[1], 0.0f);
}

// ---------------------------------------------------------------------------
// Kernel 3: segment max via unsigned atomicMax (valid: relu scores are >= 0,
// so IEEE float order == unsigned-int bit order; init value 0u == 0.0f).
// ---------------------------------------------------------------------------
__global__ void edge_max_kernel(const long long* __restrict__ ei,
                                const float* __restrict__ p_row,
                                const float* __restrict__ p_col,
                                const float* __restrict__ b,
                                unsigned int* __restrict__ seg_max_u,
                                int num_edges) {
    int e = blockIdx.x * blockDim.x + threadIdx.x;
    if (e >= num_edges) return;
    int r; float s0, s1;
    edge_scores(ei, p_row, p_col, b, e, num_edges, r, s0, s1);
    atomicMax(&seg_max_u[r * 2 + 0], __float_as_uint(s0));
    atomicMax(&seg_max_u[r * 2 + 1], __float_as_uint(s1));
}

// ---------------------------------------------------------------------------
// Kernel 4: segment sum of exp(s - seg_max) via float atomicAdd
// (lowered to GLOBAL_ATOMIC_ADD_F32, resolved in L2).
// ---------------------------------------------------------------------------
__global__ void edge_sum_kernel(const long long* __restrict__ ei,
                                const float* __restrict__ p_row,
                                const float* __restrict__ p_col,
                                const float* __restrict__ b,
                                const float* __restrict__ seg_max,
                                float* __restrict__ seg_sum,
                                int num_edges) {
    int e = blockIdx.x * blockDim.x + threadIdx.x;
    if (e >= num_edges) return;
    int r; float s0, s1;
    edge_scores(ei, p_row, p_col, b, e, num_edges, r, s0, s1);
    const float e0 = __expf(s0 - seg_max[r * 2 + 0]);
    const float e1 = __expf(s1 - seg_max[r * 2 + 1]);
    atomicAdd(&seg_sum[r * 2 + 0], e0);
    atomicAdd(&seg_sum[r * 2 + 1], e1);
}

// ---------------------------------------------------------------------------
// Kernel 5: normalized softmax, mask by actual_amount, sum the 2 channels.
// ---------------------------------------------------------------------------
__global__ void edge_out_kernel(const long long* __restrict__ ei,
                                const float* __restrict__ p_row,
                                const float* __restrict__ p_col,
                                const float* __restrict__ b,
                                const float* __restrict__ seg_max,
                                const float* __restrict__ seg_sum,
                                const int* __restrict__ amount,
                                float* __restrict__ out,
                                int num_edges) {
    int e = blockIdx.x * blockDim.x + threadIdx.x;
    if (e >= num_edges) return;
    float result = 0.0f;
    if (amount[e] != 0) {
        int r; float s0, s1;
        edge_scores(ei, p_row, p_col, b, e, num_edges, r, s0, s1);
        const float e0 = __expf(s0 - seg_max[r * 2 + 0]);
        const float e1 = __expf(s1 - seg_max[r * 2 + 1]);
        result = e0 / (seg_sum[r * 2 + 0] + EPS) + e1 / (seg_sum[r * 2 + 1] + EPS);
    }
    out[e] = result;
}

// ---------------------------------------------------------------------------
// Launch
// ---------------------------------------------------------------------------
extern "C" void kernel_launch(void* const* d_in, const int* in_sizes, int n_in,
                              void* d_out, int out_size, void* d_ws, size_t ws_size,
                              hipStream_t stream) {
    const float*     x      = (const float*)d_in[0];      // [N, 64] fp32
    const long long* ei     = (const long long*)d_in[1];  // [2, E] int64
    const int*       amount = (const int*)d_in[2];        // [E, 1] int32
    const float*     W      = (const float*)d_in[3];      // [2, 128] fp32
    const float*     b      = (const float*)d_in[4];      // [2] fp32

    const int n_nodes   = in_sizes[0] / NODE_DIM;         // 50000
    const int num_edges = out_size;                       // 1600000
    float* out = (float*)d_out;

    // Workspace: p_row | p_col | seg_max | seg_sum, each n_nodes*2 floats (1.6 MB).
    float* p_row   = (float*)d_ws;
    float* p_col   = p_row   + 2 * (size_t)n_nodes;
    float* seg_max = p_col   + 2 * (size_t)n_nodes;
    float* seg_sum = seg_max + 2 * (size_t)n_nodes;

    // 1) init seg_max + seg_sum (contiguous 4*n floats)
    {
        int cnt = 4 * n_nodes;
        init_seg_kernel<<<(cnt + 255) / 256, 256, 0, stream>>>(seg_max, cnt);
    }

    // 2) node projections via fp32 WMMA: one wave per 16-node tile, 4 waves/block
    {
        int tiles  = (n_nodes + 15) / 16;
        int blocks = (tiles + 3) / 4;
        node_proj_wmma_kernel<<<blocks, 128, 0, stream>>>(x, W, p_row, p_col, n_nodes);
    }

    // 3-5) edge passes
    {
        int blocks = (num_edges + 255) / 256;
        edge_max_kernel<<<blocks, 256, 0, stream>>>(ei, p_row, p_col, b,
                                                    (unsigned int*)seg_max, num_edges);
        edge_sum_kernel<<<blocks, 256, 0, stream>>>(ei, p_row, p_col, b,
                                                    seg_max, seg_sum, num_edges);
        edge_out_kernel<<<blocks, 256, 0, stream>>>(ei, p_row, p_col, b,
                                                    seg_max, seg_sum, amount,
                                                    out, num_edges);
    }
}